// GlobalFilterLayer_15925738733974
// MI455X (gfx1250) — compile-verified
//
#include <hip/hip_runtime.h>

// GlobalFilterLayer: Y = real(IFFT2(FFT2(X) * filt)) for B=16, C=64, H=W=256.
// Dense DFT-matrix GEMMs on fp32 matrix cores (V_WMMA_F32_16X16X4_F32).
// One 256-thread workgroup per image (persistent), 64x128 block tile,
// 2x2 register blocking per wave (32x32 output patch per wave).

#define N 256
#define NN 65536            // 256*256
#define LDA 34              // padded LDS stride (floats): 8B-aligned, conflict-free

typedef __attribute__((ext_vector_type(2))) float v2f;
typedef __attribute__((ext_vector_type(8))) float v8f;

static __device__ __forceinline__ v8f wmma4(v2f a, v2f b, v8f c) {
  // D = A(16x4) * B(4x16) + C(16x16), fp32 in / fp32 accumulate
  return __builtin_amdgcn_wmma_f32_16x16x4_f32(
      /*neg_a=*/false, a, /*neg_b=*/false, b,
      /*c_mod=*/(short)0, c, /*reuse_a=*/false, /*reuse_b=*/false);
}

// One full 256x256x256 (complex-combination) GEMM done by one 256-thread WG.
// STAGE 0: C(cplx) = A(cplx) * B(real)                     [G = F X]
// STAGE 1: C(cplx) = (A(cplx) * B(cplx)) * filt            [H = (G F) . filt]
// STAGE 2: C(cplx) = conj(A)(cplx) * B(cplx)               [K = conj(F) H]
// STAGE 3: C(real) = real(A(cplx) * conj(B)(cplx)) / N^2   [Y = real(K conj(F))/N^2]
template <int STAGE>
static __device__ __forceinline__ void gemm_stage(
    const float* __restrict__ Are, const float* __restrict__ Aim,
    const float* __restrict__ Bre, const float* __restrict__ Bim,
    float* __restrict__ Cre, float* __restrict__ Cim,
    const float* __restrict__ filt,
    float* As_re, float* As_im, float* Bs_re, float* Bs_im) {
  const int tid  = threadIdx.x;
  const int lane = tid & 31;
  const int w    = tid >> 5;       // wave id 0..7
  const int wr   = w >> 2;         // 0..1  (32-row patch within 64-row block)
  const int wc   = w & 3;          // 0..3  (32-col patch within 128-col block)
  const int l15  = lane & 15;
  const int hi   = lane >> 4;      // 0 or 1
  const int koff = 2 * hi;         // lanes 16..31 hold K+2,K+3

  constexpr bool HAS_BIM = (STAGE != 0);

  const int aRow0 = (wr * 32 + l15) * LDA + koff;
  const int aRow1 = aRow0 + 16 * LDA;
  const int bRow0 = (wc * 32 + l15) * LDA + koff;
  const int bRow1 = bRow0 + 16 * LDA;

  for (int mb = 0; mb < 4; ++mb) {
    const int m0 = mb * 64;
    for (int nb = 0; nb < 2; ++nb) {
      const int n0 = nb * 128;
      v8f accRe[2][2];
      v8f accIm[2][2];
#pragma unroll
      for (int t = 0; t < 2; ++t)
#pragma unroll
        for (int u = 0; u < 2; ++u) {
          accRe[t][u] = (v8f){0.f, 0.f, 0.f, 0.f, 0.f, 0.f, 0.f, 0.f};
          accIm[t][u] = (v8f){0.f, 0.f, 0.f, 0.f, 0.f, 0.f, 0.f, 0.f};
        }

      for (int kc = 0; kc < N; kc += 32) {
        __syncthreads();  // protect LDS from previous iteration's readers

        // ---- A panel (64 rows x 32 k) into LDS, [m][k] layout ----
        {
          const int c = (tid & 7) * 4;
#pragma unroll
          for (int h = 0; h < 2; ++h) {
            const int r = (tid >> 3) + 32 * h;
            const float4 va = *(const float4*)(Are + (size_t)(m0 + r) * N + kc + c);
            As_re[r * LDA + c + 0] = va.x;
            As_re[r * LDA + c + 1] = va.y;
            As_re[r * LDA + c + 2] = va.z;
            As_re[r * LDA + c + 3] = va.w;
            const float4 vi = *(const float4*)(Aim + (size_t)(m0 + r) * N + kc + c);
            As_im[r * LDA + c + 0] = vi.x;
            As_im[r * LDA + c + 1] = vi.y;
            As_im[r * LDA + c + 2] = vi.z;
            As_im[r * LDA + c + 3] = vi.w;
          }
        }
        // ---- B panel (32 k x 128 n) into LDS transposed, [n][k] layout ----
        {
          const int nc = (tid & 31) * 4;
#pragma unroll
          for (int h = 0; h < 4; ++h) {
            const int kr = (tid >> 5) + 8 * h;
            const float4 vb = *(const float4*)(Bre + (size_t)(kc + kr) * N + n0 + nc);
            Bs_re[(nc + 0) * LDA + kr] = vb.x;
            Bs_re[(nc + 1) * LDA + kr] = vb.y;
            Bs_re[(nc + 2) * LDA + kr] = vb.z;
            Bs_re[(nc + 3) * LDA + kr] = vb.w;
            if (HAS_BIM) {
              const float4 vc = *(const float4*)(Bim + (size_t)(kc + kr) * N + n0 + nc);
              Bs_im[(nc + 0) * LDA + kr] = vc.x;
              Bs_im[(nc + 1) * LDA + kr] = vc.y;
              Bs_im[(nc + 2) * LDA + kr] = vc.z;
              Bs_im[(nc + 3) * LDA + kr] = vc.w;
            }
          }
        }
        __syncthreads();

#pragma unroll
        for (int k = 0; k < 32; k += 4) {
          v2f ar[2], ai[2], br[2], bi[2];
          ar[0] = *(const v2f*)(As_re + aRow0 + k);
          ar[1] = *(const v2f*)(As_re + aRow1 + k);
          ai[0] = *(const v2f*)(As_im + aRow0 + k);
          ai[1] = *(const v2f*)(As_im + aRow1 + k);
          br[0] = *(const v2f*)(Bs_re + bRow0 + k);
          br[1] = *(const v2f*)(Bs_re + bRow1 + k);
          if (HAS_BIM) {
            bi[0] = *(const v2f*)(Bs_im + bRow0 + k);
            bi[1] = *(const v2f*)(Bs_im + bRow1 + k);
          }
          v2f nai[2];
          if (STAGE == 1 || STAGE == 2) {
            nai[0] = -ai[0];
            nai[1] = -ai[1];
          }
#pragma unroll
          for (int t = 0; t < 2; ++t) {
#pragma unroll
            for (int u = 0; u < 2; ++u) {
              if (STAGE == 0) {
                accRe[t][u] = wmma4(ar[t], br[u], accRe[t][u]);
                accIm[t][u] = wmma4(ai[t], br[u], accIm[t][u]);
              } else if (STAGE == 1) {      // (Are+iAim)(Bre+iBim)
                accRe[t][u] = wmma4(ar[t], br[u], accRe[t][u]);
                accRe[t][u] = wmma4(nai[t], bi[u], accRe[t][u]);
                accIm[t][u] = wmma4(ar[t], bi[u], accIm[t][u]);
                accIm[t][u] = wmma4(ai[t], br[u], accIm[t][u]);
              } else if (STAGE == 2) {      // (Are-iAim)(Bre+iBim)
                accRe[t][u] = wmma4(ar[t], br[u], accRe[t][u]);
                accRe[t][u] = wmma4(ai[t], bi[u], accRe[t][u]);
                accIm[t][u] = wmma4(ar[t], bi[u], accIm[t][u]);
                accIm[t][u] = wmma4(nai[t], br[u], accIm[t][u]);
              } else {                      // STAGE 3: real((Are+iAim)(Bre-iBim))
                accRe[t][u] = wmma4(ar[t], br[u], accRe[t][u]);
                accRe[t][u] = wmma4(ai[t], bi[u], accRe[t][u]);
              }
            }
          }
        }
      }

      // ---- epilogue: C/D layout: VGPR i -> row (i + 8*hi), col = lane&15 ----
#pragma unroll
      for (int t = 0; t < 2; ++t) {
#pragma unroll
        for (int u = 0; u < 2; ++u) {
          const int mBase = m0 + wr * 32 + t * 16 + 8 * hi;
          const int nCol  = n0 + wc * 32 + u * 16 + l15;
#pragma unroll
          for (int i = 0; i < 8; ++i) {
            const int row = mBase + i;
            const size_t idx = (size_t)row * N + nCol;
            if (STAGE == 1) {
              const float f = filt[idx];
              Cre[idx] = accRe[t][u][i] * f;
              Cim[idx] = accIm[t][u][i] * f;
            } else if (STAGE == 3) {
              Cre[idx] = accRe[t][u][i] * (1.0f / 65536.0f);
            } else {
              Cre[idx] = accRe[t][u][i];
              Cim[idx] = accIm[t][u][i];
            }
          }
        }
      }
    }
  }
}

// Build DFT twiddle tables: F[j][k] = exp(-2*pi*i*j*k/256)
__global__ void gfl_twiddle(float* __restrict__ Fre, float* __restrict__ Fim) {
  const int idx = blockIdx.x * blockDim.x + threadIdx.x;  // 0..65535
  const int j = idx >> 8, k = idx & 255;
  const int t = (j * k) & 255;  // exploit periodicity for accuracy
  const float ang = -6.283185307179586f * (float)t * (1.0f / 256.0f);
  Fre[idx] = cosf(ang);
  Fim[idx] = sinf(ang);
}

__global__ __launch_bounds__(256) void gfl_main(
    const float* __restrict__ x, const float* __restrict__ filters,
    float* __restrict__ out, const float* __restrict__ Fre,
    const float* __restrict__ Fim, float* __restrict__ imgws) {
  __shared__ float smem[(64 + 64 + 128 + 128) * LDA];
  float* As_re = smem;
  float* As_im = As_re + 64 * LDA;
  float* Bs_re = As_im + 64 * LDA;
  float* Bs_im = Bs_re + 128 * LDA;

  // Per-workgroup private scratch: two complex 256x256 buffers (ping-pong)
  float* Gre = imgws + (size_t)blockIdx.x * 4 * NN;
  float* Gim = Gre + NN;
  float* Hre = Gim + NN;
  float* Him = Hre + NN;

  for (int img = blockIdx.x; img < 16 * 64; img += gridDim.x) {
    const float* X = x + (size_t)img * NN;
    const float* f = filters + (size_t)(img & 63) * NN;  // channel = img % C
    float* Y = out + (size_t)img * NN;

    // G = F * X
    gemm_stage<0>(Fre, Fim, X, (const float*)nullptr, Gre, Gim,
                  (const float*)nullptr, As_re, As_im, Bs_re, Bs_im);
    __syncthreads();
    // H = (G * F) .* filt
    gemm_stage<1>(Gre, Gim, Fre, Fim, Hre, Him, f, As_re, As_im, Bs_re, Bs_im);
    __syncthreads();
    // K = conj(F) * H   (reuse G buffers)
    gemm_stage<2>(Fre, Fim, Hre, Him, Gre, Gim, (const float*)nullptr,
                  As_re, As_im, Bs_re, Bs_im);
    __syncthreads();
    // Y = real(K * conj(F)) / N^2
    gemm_stage<3>(Gre, Gim, Fre, Fim, Y, (float*)nullptr, (const float*)nullptr,
                  As_re, As_im, Bs_re, Bs_im);
    __syncthreads();
  }
}

extern "C" void kernel_launch(void* const* d_in, const int* in_sizes, int n_in,
                              void* d_out, int out_size, void* d_ws, size_t ws_size,
                              hipStream_t stream) {
  const float* x       = (const float*)d_in[0];  // (16, 64, 256, 256) f32
  const float* filters = (const float*)d_in[1];  // (64, 256, 256) f32
  float* out = (float*)d_out;                    // (16, 64, 256, 256) f32
  float* ws  = (float*)d_ws;

  float* Fre   = ws;            // 65536 floats
  float* Fim   = ws + NN;       // 65536 floats
  float* imgws = ws + 2 * NN;   // per-WG buffers follow

  // Each concurrent workgroup needs 4 * 256KB = 1MB of scratch.
  const size_t table_bytes  = (size_t)2 * NN * sizeof(float);
  const size_t per_wg_bytes = (size_t)4 * NN * sizeof(float);
  size_t avail = (ws_size > table_bytes) ? (ws_size - table_bytes) : 0;
  int nWG = (int)(avail / per_wg_bytes);
  if (nWG < 1) nWG = 1;
  if (nWG > 16 * 64) nWG = 16 * 64;

  gfl_twiddle<<<NN / 256, 256, 0, stream>>>(Fre, Fim);
  gfl_main<<<nWG, 256, 0, stream>>>(x, filters, out, Fre, Fim, imgws);
}